// STRGCN_23837068493035
// MI455X (gfx1250) — compile-verified
//
#include <hip/hip_runtime.h>
#include <math.h>

#define NN 50000
#define EE 800000
#define F_IN 128
#define HID 32
#define CLS 16

typedef float v2f __attribute__((ext_vector_type(2)));
typedef float v8f __attribute__((ext_vector_type(8)));

// ---------------------------------------------------------------------------
// zero scratch accumulators
// ---------------------------------------------------------------------------
__global__ void k_zero(float* __restrict__ p, int n) {
    int i = blockIdx.x * blockDim.x + threadIdx.x;
    if (i < n) p[i] = 0.0f;
}

// ---------------------------------------------------------------------------
// Layer-1 GEMMs: xr = x @ W_rel1 ; xroot = x @ W_root1 + b1   (K = 128, N = 32)
// One wave handles a 16-row tile; 4 accumulators (rel/root x 2 col-tiles).
// Uses V_WMMA_F32_16X16X4_F32.  A 16x4 layout: lanes0-15 hold K=kb,kb+1 in
// v0,v1; lanes16-31 hold K=kb+2,kb+3.  B 4x16 mirrors it (row striped across
// lanes within a VGPR).  D 16x16: vgpr r => M = r (+8 for lanes16-31), N=lane%16.
// ---------------------------------------------------------------------------
__global__ void k_gemm1(const float* __restrict__ x,
                        const float* __restrict__ Wrel,
                        const float* __restrict__ Wroot,
                        const float* __restrict__ b1,
                        float* __restrict__ xr,
                        float* __restrict__ xroot) {
    const int wave = threadIdx.x >> 5;
    const int lane = threadIdx.x & 31;
    const int tile = blockIdx.x * 4 + wave;          // 16-row tile index
    const int rowbase = tile * 16;
    if (rowbase >= NN) return;                       // wave-uniform: EXEC stays all-1s

    const int h  = lane >> 4;                        // 0: lanes 0-15, 1: lanes 16-31
    const int n  = lane & 15;
    const int arow = rowbase + n;

    v8f acc_rel0 = {}, acc_rel1 = {}, acc_root0 = {}, acc_root1 = {};

    for (int kb = 0; kb < F_IN; kb += 4) {
        const int k0 = kb + 2 * h;
        v2f a;
        a.x = x[arow * F_IN + k0];
        a.y = x[arow * F_IN + k0 + 1];

        const float* wr = Wrel  + k0 * HID;
        const float* wo = Wroot + k0 * HID;
        v2f brel0, brel1, broot0, broot1;
        brel0.x  = wr[n];            brel0.y  = wr[HID + n];
        brel1.x  = wr[16 + n];       brel1.y  = wr[HID + 16 + n];
        broot0.x = wo[n];            broot0.y = wo[HID + n];
        broot1.x = wo[16 + n];       broot1.y = wo[HID + 16 + n];

        acc_rel0  = __builtin_amdgcn_wmma_f32_16x16x4_f32(false, a, false, brel0,  (short)0, acc_rel0,  false, false);
        acc_rel1  = __builtin_amdgcn_wmma_f32_16x16x4_f32(false, a, false, brel1,  (short)0, acc_rel1,  false, false);
        acc_root0 = __builtin_amdgcn_wmma_f32_16x16x4_f32(false, a, false, broot0, (short)0, acc_root0, false, false);
        acc_root1 = __builtin_amdgcn_wmma_f32_16x16x4_f32(false, a, false, broot1, (short)0, acc_root1, false, false);
    }

    const float bias0 = b1[n];
    const float bias1 = b1[16 + n];
    #pragma unroll
    for (int r = 0; r < 8; ++r) {
        const int row = rowbase + r + 8 * h;
        xr[row * HID + n]          = acc_rel0[r];
        xr[row * HID + 16 + n]     = acc_rel1[r];
        xroot[row * HID + n]       = acc_root0[r] + bias0;
        xroot[row * HID + 16 + n]  = acc_root1[r] + bias1;
    }
}

// ---------------------------------------------------------------------------
// Edge scatter, layer 1: agg1[dst] += xr[src] * w   (32 feats/edge, L2 atomics)
// ---------------------------------------------------------------------------
__global__ void k_scatter1(const long long* __restrict__ ei,
                           const float* __restrict__ ea,
                           const float* __restrict__ xr,
                           float* __restrict__ agg1) {
    const int t = blockIdx.x * blockDim.x + threadIdx.x;
    const int e = t >> 5;
    const int f = t & 31;
    if (e >= EE) return;
    const long long s = ei[e];
    const long long d = ei[EE + e];
    const float w = ea[e];
    atomicAdd(&agg1[d * HID + f], xr[s * HID + f] * w);
}

// ---------------------------------------------------------------------------
// Layer-2 GEMMs with fused ReLU:  A = relu(agg1 + xroot)  (never materialized)
// hr = A @ W_rel2 ; hroot = A @ W_root2 + b2    (K = 32, N = 16)
// ---------------------------------------------------------------------------
__global__ void k_gemm2(const float* __restrict__ agg1,
                        const float* __restrict__ xroot,
                        const float* __restrict__ Wrel2,
                        const float* __restrict__ Wroot2,
                        const float* __restrict__ b2,
                        float* __restrict__ hr,
                        float* __restrict__ hroot) {
    const int wave = threadIdx.x >> 5;
    const int lane = threadIdx.x & 31;
    const int tile = blockIdx.x * 4 + wave;
    const int rowbase = tile * 16;
    if (rowbase >= NN) return;

    const int h = lane >> 4;
    const int n = lane & 15;
    const int arow = rowbase + n;

    v8f acc_rel = {}, acc_root = {};

    for (int kb = 0; kb < HID; kb += 4) {
        const int k0 = kb + 2 * h;
        v2f a;
        a.x = fmaxf(agg1[arow * HID + k0]     + xroot[arow * HID + k0],     0.0f);
        a.y = fmaxf(agg1[arow * HID + k0 + 1] + xroot[arow * HID + k0 + 1], 0.0f);

        v2f brel, broot;
        brel.x  = Wrel2[k0 * CLS + n];        brel.y  = Wrel2[(k0 + 1) * CLS + n];
        broot.x = Wroot2[k0 * CLS + n];       broot.y = Wroot2[(k0 + 1) * CLS + n];

        acc_rel  = __builtin_amdgcn_wmma_f32_16x16x4_f32(false, a, false, brel,  (short)0, acc_rel,  false, false);
        acc_root = __builtin_amdgcn_wmma_f32_16x16x4_f32(false, a, false, broot, (short)0, acc_root, false, false);
    }

    const float bias = b2[n];
    #pragma unroll
    for (int r = 0; r < 8; ++r) {
        const int row = rowbase + r + 8 * h;
        hr[row * CLS + n]    = acc_rel[r];
        hroot[row * CLS + n] = acc_root[r] + bias;
    }
}

// ---------------------------------------------------------------------------
// Edge scatter, layer 2: agg2[dst] += hr[src] * w   (16 feats/edge)
// ---------------------------------------------------------------------------
__global__ void k_scatter2(const long long* __restrict__ ei,
                           const float* __restrict__ ea,
                           const float* __restrict__ hr,
                           float* __restrict__ agg2) {
    const int t = blockIdx.x * blockDim.x + threadIdx.x;
    const int e = t >> 4;
    const int f = t & 15;
    if (e >= EE) return;
    const long long s = ei[e];
    const long long d = ei[EE + e];
    const float w = ea[e];
    atomicAdd(&agg2[d * CLS + f], hr[s * CLS + f] * w);
}

// ---------------------------------------------------------------------------
// out = log_softmax(agg2 + hroot) over 16 classes
// ---------------------------------------------------------------------------
__global__ void k_logsoftmax(const float* __restrict__ agg2,
                             const float* __restrict__ hroot,
                             float* __restrict__ out) {
    const int row = blockIdx.x * blockDim.x + threadIdx.x;
    if (row >= NN) return;
    float v[CLS];
    float m = -INFINITY;
    #pragma unroll
    for (int i = 0; i < CLS; ++i) {
        v[i] = agg2[row * CLS + i] + hroot[row * CLS + i];
        m = fmaxf(m, v[i]);
    }
    float s = 0.0f;
    #pragma unroll
    for (int i = 0; i < CLS; ++i) s += __expf(v[i] - m);
    const float lse = m + __logf(s);
    #pragma unroll
    for (int i = 0; i < CLS; ++i) out[row * CLS + i] = v[i] - lse;
}

// ---------------------------------------------------------------------------
extern "C" void kernel_launch(void* const* d_in, const int* in_sizes, int n_in,
                              void* d_out, int out_size, void* d_ws, size_t ws_size,
                              hipStream_t stream) {
    const float*     x      = (const float*)d_in[0];
    const long long* ei     = (const long long*)d_in[1];   // int64 [2, E]
    const float*     ea     = (const float*)d_in[2];
    const float*     Wrel1  = (const float*)d_in[3];
    const float*     Wroot1 = (const float*)d_in[4];
    const float*     b1     = (const float*)d_in[5];
    const float*     Wrel2  = (const float*)d_in[6];
    const float*     Wroot2 = (const float*)d_in[7];
    const float*     b2     = (const float*)d_in[8];
    float*           out    = (float*)d_out;

    float* ws    = (float*)d_ws;
    float* xr    = ws;                       // [N,32]
    float* xroot = xr    + (size_t)NN * HID; // [N,32]
    float* agg1  = xroot + (size_t)NN * HID; // [N,32]
    float* hr    = agg1  + (size_t)NN * HID; // [N,16]
    float* hroot = hr    + (size_t)NN * CLS; // [N,16]
    float* agg2  = hroot + (size_t)NN * CLS; // [N,16]

    // zero the span [agg1 .. agg2+N*CLS): covers agg1, hr, hroot, agg2
    const int nzero = NN * HID + 3 * NN * CLS;
    k_zero<<<(nzero + 255) / 256, 256, 0, stream>>>(agg1, nzero);

    const int ntiles = NN / 16;                         // 3125
    const int gblocks = (ntiles + 3) / 4;               // 4 waves (tiles) per block

    k_gemm1<<<gblocks, 128, 0, stream>>>(x, Wrel1, Wroot1, b1, xr, xroot);

    k_scatter1<<<(EE * 32) / 256, 256, 0, stream>>>(ei, ea, xr, agg1);

    k_gemm2<<<gblocks, 128, 0, stream>>>(agg1, xroot, Wrel2, Wroot2, b2, hr, hroot);

    k_scatter2<<<(EE * 16) / 256, 256, 0, stream>>>(ei, ea, hr, agg2);

    k_logsoftmax<<<(NN + 255) / 256, 256, 0, stream>>>(agg2, hroot, out);
}